// SpatialTransformer_27410481283460
// MI455X (gfx1250) — compile-verified
//
#include <hip/hip_runtime.h>
#include <hip/hip_bf16.h>

// Fixed problem shape from reference setup_inputs():
//   src  [B=2, C=4, D=128, H=160, W=160] f32
//   flow [B=2, 3,   D,     H,     W    ] f32
//   out  [B, C, D, H, W] f32
constexpr int Bn = 2;
constexpr int Cn = 4;
constexpr int Dn = 128;
constexpr int Hn = 160;
constexpr int Wn = 160;
constexpr int DHW = Dn * Hn * Wn;              // 3,276,800
constexpr long long NTOT = (long long)Bn * DHW; // 6,553,600 (= 25600 * 256)

__global__ __launch_bounds__(256)
void warp3d_trilinear_kernel(const float* __restrict__ src,
                             const float* __restrict__ flow,
                             float* __restrict__ out)
{
    const int i = blockIdx.x * 256 + threadIdx.x;
    if (i >= (int)NTOT) return;

    // Decompose flat index -> (b, z, y, x); s = spatial index within batch.
    const int b = i / DHW;
    const int s = i - b * DHW;
    const int x = s % Wn;
    const int t = s / Wn;
    const int y = t % Hn;
    const int z = t / Hn;

    // Flow components live in separate planes: [b, comp, z, y, x].
    // Streamed exactly once -> non-temporal so they don't evict src from L2.
    const float* fb = flow + (size_t)b * 3 * DHW + s;
    const float fz = __builtin_nontemporal_load(fb);
    const float fy = __builtin_nontemporal_load(fb + DHW);
    const float fx = __builtin_nontemporal_load(fb + 2 * DHW);

    // Sampling coordinate in voxel units (align_corners=True semantics).
    const float zc = (float)z + fz;
    const float yc = (float)y + fy;
    const float xc = (float)x + fx;

    const float z0f = floorf(zc);
    const float y0f = floorf(yc);
    const float x0f = floorf(xc);
    const float wz = zc - z0f;
    const float wy = yc - y0f;
    const float wx = xc - x0f;
    const int z0 = (int)z0f;
    const int y0 = (int)y0f;
    const int x0 = (int)x0f;

    const float wzv[2] = {1.0f - wz, wz};
    const float wyv[2] = {1.0f - wy, wy};
    const float wxv[2] = {1.0f - wx, wx};

    // src base for this batch; channel stride is DHW.
    const float* sb = src + (size_t)b * Cn * DHW;

    float acc0 = 0.0f, acc1 = 0.0f, acc2 = 0.0f, acc3 = 0.0f;

#pragma unroll
    for (int dz = 0; dz < 2; ++dz) {
        const int  zi = z0 + dz;
        const bool vz = (unsigned)zi < (unsigned)Dn;
        const int  zcl = min(max(zi, 0), Dn - 1);
#pragma unroll
        for (int dy = 0; dy < 2; ++dy) {
            const int  yi = y0 + dy;
            const bool vy = (unsigned)yi < (unsigned)Hn;
            const int  ycl = min(max(yi, 0), Hn - 1);
            const float wzy   = wzv[dz] * wyv[dy];
            const int   bzy   = (zcl * Hn + ycl) * Wn;
            const bool  vzy   = vz && vy;
#pragma unroll
            for (int dx = 0; dx < 2; ++dx) {
                const int  xi = x0 + dx;
                const bool vx = (unsigned)xi < (unsigned)Wn;
                const int  xcl = min(max(xi, 0), Wn - 1);
                // zeros padding: fold validity into the weight (one cndmask
                // per corner); address is clamped so the load is always legal.
                const float w = (vzy && vx) ? (wzy * wxv[dx]) : 0.0f;
                const int idx = bzy + xcl;
                // 4 channels share one address computation; src stays RT so
                // the whole 105 MB volume remains resident in the 192 MB L2.
                acc0 = fmaf(w, sb[idx            ], acc0);
                acc1 = fmaf(w, sb[idx +     DHW  ], acc1);
                acc2 = fmaf(w, sb[idx + 2 * DHW  ], acc2);
                acc3 = fmaf(w, sb[idx + 3 * DHW  ], acc3);
            }
        }
    }

    // Output is write-once streaming -> non-temporal stores.
    float* ob = out + (size_t)b * Cn * DHW + s;
    __builtin_nontemporal_store(acc0, ob);
    __builtin_nontemporal_store(acc1, ob + DHW);
    __builtin_nontemporal_store(acc2, ob + 2 * DHW);
    __builtin_nontemporal_store(acc3, ob + 3 * DHW);
}

extern "C" void kernel_launch(void* const* d_in, const int* in_sizes, int n_in,
                              void* d_out, int out_size, void* d_ws, size_t ws_size,
                              hipStream_t stream)
{
    (void)in_sizes; (void)n_in; (void)d_ws; (void)ws_size; (void)out_size;
    const float* src  = (const float*)d_in[0];
    const float* flow = (const float*)d_in[1];
    float*       out  = (float*)d_out;

    const int threads = 256;                       // 8 wave32 waves per block
    const int blocks  = (int)(NTOT / threads);     // exact: 25,600
    warp3d_trilinear_kernel<<<blocks, threads, 0, stream>>>(src, flow, out);
}